// MultiPromptCLIP_36077725287058
// MI455X (gfx1250) — compile-verified
//
#include <hip/hip_runtime.h>
#include <hip/hip_bf16.h>
#include <limits.h>

// Problem constants from the reference.
#define B_  512
#define C_  6
#define T_  77
#define D_  512
#define LI_ 193

#define MROWS 16   // padded M (C=6 -> 16)
#define MCOLS 80   // padded K (T=77 -> 80)
#define MPITCH 84  // LDS row pitch in floats (bank-conflict-free for 16 rows)

typedef __attribute__((ext_vector_type(2))) float v2f;
typedef __attribute__((ext_vector_type(8))) float v8f;

// ---------------------------------------------------------------------------
// Kernel 1: local_text[b,c,d] = sum_t mask[b,c,t] * text[b,t,d] / 77
// One wave32 per (batch, 16-wide N tile). M=6 padded to 16, K=77 padded to 80.
// V_WMMA_F32_16X16X4_F32, branch-free inner loop (A from zero-padded LDS,
// B via base+immediate-offset global loads, select-based K tail).
// ---------------------------------------------------------------------------
__global__ __launch_bounds__(128) void local_text_wmma_kernel(
    const float* __restrict__ text,   // [B, T, D]
    const int*   __restrict__ mask,   // [B, C, T]
    float*       __restrict__ out_lt) // [B, C, D]
{
    __shared__ float smask[MROWS * MPITCH];

    const int tid   = threadIdx.x;
    const int lane  = tid & 31;
    const int wave  = blockIdx.x * 4 + (tid >> 5);
    const int b     = wave >> 5;          // 32 N-tiles per batch; all 4 waves
    const int ntile = wave & 31;          // in a block share one batch
    const int half  = lane >> 4;          // 0: lanes 0-15, 1: lanes 16-31
    const int l     = lane & 15;          // A/D row, or B column
    const int ncol  = ntile * 16 + l;

    // --- Stage mask -> LDS as zero-padded 16x80 floats (one-time, coalesced).
    for (int idx = tid; idx < MROWS * MPITCH; idx += 128) smask[idx] = 0.0f;
    __syncthreads();
    const int* __restrict__ mk = mask + (size_t)b * C_ * T_;
    for (int idx = tid; idx < C_ * T_; idx += 128) {
        const int row = idx / T_;
        const int col = idx - row * T_;
        smask[row * MPITCH + col] = (float)mk[idx];
    }
    __syncthreads();

    // Per-lane base pointers: all inner-loop accesses are immediate offsets.
    // A layout (16x4 f32): lanes 0-15 -> K=k0,k0+1 ; lanes 16-31 -> k0+2,k0+3.
    const float* __restrict__ abase = &smask[l * MPITCH + half * 2];
    const float* __restrict__ tbase =
        text + (size_t)b * T_ * D_ + (size_t)(half * 2) * D_ + ncol;

    v8f acc = {};

    // Main K loop: t = 0..75, no bounds checks needed (straight-line body).
#pragma unroll
    for (int k0 = 0; k0 < 76; k0 += 4) {
        v2f a;
        a.x = abase[k0];               // ds_load (pairs merge to 2addr_b64)
        a.y = abase[k0 + 1];
        v2f bt;
        bt.x = tbase[(size_t)k0 * D_];         // global_load offset:k0*2048
        bt.y = tbase[(size_t)(k0 + 1) * D_];
        acc = __builtin_amdgcn_wmma_f32_16x16x4_f32(
            false, a, false, bt, (short)0, acc, false, false);
    }

    // Tail step (t = 76..79): only t=76 is real. Load row 76 unconditionally
    // (in-bounds for every lane) and select-zero the padded K slots.
    {
        v2f a;
        a.x = abase[76];               // LDS already zero for t >= 77
        a.y = abase[77];
        const float v76 = text[(size_t)b * T_ * D_ + (size_t)76 * D_ + ncol];
        v2f bt;
        bt.x = half ? 0.0f : v76;      // branch-free v_cndmask
        bt.y = 0.0f;
        acc = __builtin_amdgcn_wmma_f32_16x16x4_f32(
            false, a, false, bt, (short)0, acc, false, false);
    }

    // D layout: VGPR r on lanes 0-15 holds row M=r; lanes 16-31 hold M=r+8
    // (>= C here), so only the low half stores. Scale by reciprocal instead of
    // dividing: avoids the 8-op IEEE div expansion per element.
    if (half == 0) {
        const float inv_t = 1.0f / (float)T_;   // compile-time constant
#pragma unroll
        for (int r = 0; r < C_; ++r) {
            out_lt[((size_t)b * C_ + r) * D_ + ncol] = acc[r] * inv_t;
        }
    }
}

// ---------------------------------------------------------------------------
// Kernel 2: per-batch argmax over captions (first max, like jnp.argmax),
// gather global_text row, copy global_image row 0, pass logit_scale.
// One 128-thread block per batch; 512 floats per row -> one float4 per thread.
// ---------------------------------------------------------------------------
__global__ __launch_bounds__(128) void gather_copy_kernel(
    const float* __restrict__ img,      // [B, LI, D]
    const float* __restrict__ text,     // [B, T, D]
    const float* __restrict__ lscale,   // [1]
    const int*   __restrict__ captions, // [B, T]
    float*       __restrict__ out)      // concatenated outputs
{
    __shared__ int sval[128];
    __shared__ int sidx[128];

    const int b = blockIdx.x;
    const int t = threadIdx.x;

    // Argmax with first-occurrence tie-break.
    sval[t] = (t < T_) ? captions[(size_t)b * T_ + t] : INT_MIN;
    sidx[t] = t;
    __syncthreads();
#pragma unroll
    for (int s = 64; s > 0; s >>= 1) {
        if (t < s) {
            const int v2 = sval[t + s];
            const int i2 = sidx[t + s];
            if (v2 > sval[t] || (v2 == sval[t] && i2 < sidx[t])) {
                sval[t] = v2;
                sidx[t] = i2;
            }
        }
        __syncthreads();
    }
    const int eot = sidx[0];

    // Output slab offsets (flat, return order).
    float* out_gt = out;                                    // [B, D]
    float* out_gi = out + (size_t)B_ * D_;                  // [B, D]
    float* out_ls = out + 2 * (size_t)B_ * D_
                        + (size_t)B_ * C_ * D_;             // [1]

    const float4* gt_src =
        (const float4*)(text + ((size_t)b * T_ + eot) * D_);
    const float4* gi_src =
        (const float4*)(img + (size_t)b * LI_ * D_);        // row LI=0
    float4* gt_dst = (float4*)(out_gt + (size_t)b * D_);
    float4* gi_dst = (float4*)(out_gi + (size_t)b * D_);

    // D_/4 = 128 float4's == blockDim.x
    gt_dst[t] = gt_src[t];
    gi_dst[t] = gi_src[t];

    if (b == 0 && t == 0) out_ls[0] = lscale[0];
}

// ---------------------------------------------------------------------------
extern "C" void kernel_launch(void* const* d_in, const int* in_sizes, int n_in,
                              void* d_out, int out_size, void* d_ws, size_t ws_size,
                              hipStream_t stream) {
    (void)in_sizes; (void)n_in; (void)out_size; (void)d_ws; (void)ws_size;

    const float* img    = (const float*)d_in[0]; // image_features [B,LI,D]
    const float* text   = (const float*)d_in[1]; // text_features  [B,T,D]
    const float* lscale = (const float*)d_in[2]; // logit_scale    [1]
    const int*   caps   = (const int*)d_in[3];   // captions       [B,T]
    const int*   mask   = (const int*)d_in[4];   // noun_chunk_mask[B,C,T]

    float* out    = (float*)d_out;
    float* out_lt = out + 2 * (size_t)B_ * D_;   // local_text slab

    // 512 batches * 32 N-tiles = 16384 waves; 4 waves (128 threads) per block.
    local_text_wmma_kernel<<<(B_ * 32) / 4, 128, 0, stream>>>(text, mask, out_lt);
    gather_copy_kernel<<<B_, 128, 0, stream>>>(img, text, lscale, caps, out);
}